// AutoRNN_88587995447572
// MI455X (gfx1250) — compile-verified
//
#include <hip/hip_runtime.h>
#include <hip/hip_bf16.h>
#include <math.h>

typedef __attribute__((ext_vector_type(2))) float v2f;
typedef __attribute__((ext_vector_type(8))) float v8f;

#define T_SEQ  16384
#define DMODEL 1024
#define RNN_NWG 64

#if defined(__HIP_DEVICE_COMPILE__)
#if !__has_builtin(__builtin_amdgcn_wmma_f32_16x16x4_f32)
#error "missing __builtin_amdgcn_wmma_f32_16x16x4_f32 on this toolchain"
#endif
#endif

__device__ __forceinline__ v8f wmma_f32(v2f a, v2f b, v8f c) {
    // full-precision f32 matrix op: D = A(16x4) * B(4x16) + C(16x16)
    return __builtin_amdgcn_wmma_f32_16x16x4_f32(
        /*neg_a=*/false, a, /*neg_b=*/false, b,
        /*c_mod=*/(short)0, c, /*reuse_a=*/false, /*reuse_b=*/false);
}

// CDNA5 async global->LDS copies (ASYNCcnt-tracked, per-lane LDS dest VGPR).
// GV addressing: vaddr = full 64-bit address VGPR pair, saddr = off.
__device__ __forceinline__ void async_copy_b128(unsigned lds_off, const float* g) {
    asm volatile("global_load_async_to_lds_b128 %0, %1, off"
                 :: "v"(lds_off), "v"(g) : "memory");
}
__device__ __forceinline__ void async_copy_b32(unsigned lds_off, const float* g) {
    asm volatile("global_load_async_to_lds_b32 %0, %1, off"
                 :: "v"(lds_off), "v"(g) : "memory");
}
__device__ __forceinline__ void wait_async0() {
    asm volatile("s_wait_asynccnt 0x0" ::: "memory");
}

// ---------------------------------------------------------------------------
// C[M,N] = A[M,K] @ B[K,N] + bias[N], all row-major f32.
// BM=128 BN=64 BK=16; 256 threads = 8 waves (4 row-waves x 2 col-waves);
// each wave owns a 32x32 region = 2x2 WMMA f32 16x16 accumulators.
// Double-buffered LDS tiles staged with GLOBAL_LOAD_ASYNC_TO_LDS (fetch of
// tile k+1 overlaps WMMA on tile k; no loadcnt stalls in the staging path).
// LDS rows padded to 20 floats -> conflict-free ds_load_b64 fragment reads;
// the async-b32 scatter performs the B transpose for free.
// No divergence anywhere: EXEC is all-ones as WMMA requires.
// ---------------------------------------------------------------------------
__global__ __launch_bounds__(256) void gemm_bias_f32_wmma(
        const float* __restrict__ A, const float* __restrict__ B,
        const float* __restrict__ bias, float* __restrict__ C,
        int M, int N, int K)
{
    __shared__ float As[2][128][20];   // A tile, [m][k], 80B row stride
    __shared__ float BsT[2][64][20];   // B tile transposed, [n][k]

    const int tid  = threadIdx.x;
    const int lane = tid & 31;
    const int wid  = tid >> 5;
    const int m0 = blockIdx.y * 128;
    const int n0 = blockIdx.x * 64;

    const int rowbase = (wid & 3) * 32;   // 4 row-waves cover 128 rows
    const int colbase = (wid >> 2) * 32;  // 2 col-waves cover 64 cols
    const int r    = lane & 15;           // A row / B col within 16x16 tile
    const int koff = (lane >> 4) * 2;     // K pair select per ISA layout

    // LDS byte offsets (flat LDS addresses: low 32 bits = LDS offset)
    const unsigned asb = (unsigned)(uintptr_t)&As[0][0][0];
    const unsigned bsb = (unsigned)(uintptr_t)&BsT[0][0][0];

    v8f acc[2][2];
    acc[0][0] = {}; acc[0][1] = {}; acc[1][0] = {}; acc[1][1] = {};

    const int rowA = tid >> 2;            // A stage: 2x b128 per thread
    const int chA  = tid & 3;             // 16B chunk within a 16-float row

    auto stage = [&](int k0, int buf) {
        const unsigned abuf = asb + (unsigned)buf * 128u * 80u;
        const unsigned bbuf = bsb + (unsigned)buf * 64u * 80u;
        async_copy_b128(abuf + (unsigned)(rowA * 80 + chA * 16),
                        &A[(size_t)(m0 + rowA) * K + k0 + chA * 4]);
        async_copy_b128(abuf + (unsigned)((rowA + 64) * 80 + chA * 16),
                        &A[(size_t)(m0 + rowA + 64) * K + k0 + chA * 4]);
        #pragma unroll
        for (int j = 0; j < 4; ++j) {     // B: coalesced reads, transposed scatter
            const int e  = j * 256 + tid; // e in [0,1024): kB = e/64, nB = e%64
            const int kB = e >> 6;
            const int nB = e & 63;
            async_copy_b32(bbuf + (unsigned)(nB * 80 + kB * 4),
                           &B[(size_t)(k0 + kB) * N + n0 + nB]);
        }
    };

    stage(0, 0);
    const int nk = K / 16;
    for (int ik = 0; ik < nk; ++ik) {
        const int buf = ik & 1;
        wait_async0();        // my wave's async writes for tile ik are in LDS
        __syncthreads();      // all waves' writes done; all done reading buf^1
        if (ik + 1 < nk) stage((ik + 1) * 16, buf ^ 1);

        #pragma unroll
        for (int kk = 0; kk < 16; kk += 4) {
            v2f a0 = *(const v2f*)&As[buf][rowbase      + r][kk + koff];
            v2f a1 = *(const v2f*)&As[buf][rowbase + 16 + r][kk + koff];
            v2f b0 = *(const v2f*)&BsT[buf][colbase      + r][kk + koff];
            v2f b1 = *(const v2f*)&BsT[buf][colbase + 16 + r][kk + koff];
            acc[0][0] = wmma_f32(a0, b0, acc[0][0]);
            acc[0][1] = wmma_f32(a0, b1, acc[0][1]);
            acc[1][0] = wmma_f32(a1, b0, acc[1][0]);
            acc[1][1] = wmma_f32(a1, b1, acc[1][1]);
        }
    }

    const int hi = (lane >> 4) * 8;  // C/D layout: VGPR e holds M=e / M=e+8
    #pragma unroll
    for (int j = 0; j < 2; ++j) {
        const int col = n0 + colbase + j * 16 + r;
        const float bcol = bias[col];
        #pragma unroll
        for (int i = 0; i < 2; ++i) {
            #pragma unroll
            for (int e = 0; e < 8; ++e) {
                const int row = m0 + rowbase + i * 16 + hi + e;
                C[(size_t)row * N + col] = acc[i][j][e] + bcol;
            }
        }
    }
}

// ---------------------------------------------------------------------------
// Sequential recurrence: h_t = tanh(Pre[t] + h_{t-1} @ Wh), t = 0..T-1
// (Pre already contains x_t@Wx + bh; Wh term skipped at t=0 <=> h_{-1}=0).
// 64 persistent WGs x 256 threads. WG b owns columns [16b,16b+16);
// thread (kpart=tid>>4, col=tid&15) accumulates a 64-element K slice, then a
// 16-way LDS reduction finishes each column. Wh (4 MB) stays L2-resident.
// Cross-step coherence: h read with agent-scope atomic loads; device-wide
// generation barrier with acq/rel agent atomics in d_ws.
// ---------------------------------------------------------------------------
__global__ __launch_bounds__(256) void rnn_recurrence_kernel(
        const float* __restrict__ Wh, const float* __restrict__ Pre,
        float* __restrict__ H, unsigned* __restrict__ bar)
{
    __shared__ float red[16][17];

    const int tid       = threadIdx.x;
    const int col_local = tid & 15;
    const int kpart     = tid >> 4;              // 0..15
    const int wg        = blockIdx.x;            // 0..RNN_NWG-1
    const int col       = wg * 16 + col_local;
    const int kbase     = kpart * 64;

    unsigned gen_target = 0;
    for (int t = 0; t < T_SEQ; ++t) {
        float acc = 0.0f;
        if (t > 0) {
            const float* hprev = H + (size_t)(t - 1) * DMODEL + kbase;
            const float* wcol  = Wh + (size_t)kbase * DMODEL + col;
            #pragma unroll 8
            for (int j = 0; j < 64; ++j) {
                float hv = __hip_atomic_load(hprev + j, __ATOMIC_RELAXED,
                                             __HIP_MEMORY_SCOPE_AGENT);
                acc = fmaf(hv, wcol[(size_t)j * DMODEL], acc);
            }
        }
        red[kpart][col_local] = acc;
        __syncthreads();
        if (tid < 16) {
            float s = 0.0f;
            #pragma unroll
            for (int p = 0; p < 16; ++p) s += red[p][tid];
            s += Pre[(size_t)t * DMODEL + wg * 16 + tid];
            H[(size_t)t * DMODEL + wg * 16 + tid] = tanhf(s);
        }
        __syncthreads();

        // device-wide generation barrier (monotone gen, memset-0 init)
        ++gen_target;
        if (tid == 0) {
            unsigned arr = __hip_atomic_fetch_add(&bar[0], 1u, __ATOMIC_ACQ_REL,
                                                  __HIP_MEMORY_SCOPE_AGENT);
            if (arr == RNN_NWG - 1) {
                __hip_atomic_store(&bar[0], 0u, __ATOMIC_RELAXED,
                                   __HIP_MEMORY_SCOPE_AGENT);
                __hip_atomic_fetch_add(&bar[1], 1u, __ATOMIC_RELEASE,
                                       __HIP_MEMORY_SCOPE_AGENT);
            } else {
                while (__hip_atomic_load(&bar[1], __ATOMIC_ACQUIRE,
                                         __HIP_MEMORY_SCOPE_AGENT) < gen_target) {
                    __builtin_amdgcn_s_sleep(1);
                }
            }
        }
        __syncthreads();
    }
}

// ---------------------------------------------------------------------------
extern "C" void kernel_launch(void* const* d_in, const int* in_sizes, int n_in,
                              void* d_out, int out_size, void* d_ws, size_t ws_size,
                              hipStream_t stream) {
    (void)in_sizes; (void)n_in; (void)out_size; (void)ws_size;
    const float* X  = (const float*)d_in[0];   // [T, D]
    const float* Wx = (const float*)d_in[1];   // [D, D]
    const float* Wh = (const float*)d_in[2];   // [D, D]
    const float* Wy = (const float*)d_in[3];   // [D, D]
    const float* bh = (const float*)d_in[4];   // [D]
    const float* by = (const float*)d_in[5];   // [D]

    float* Hout = (float*)d_out;                          // hidden  [T, D]
    float* Yout = (float*)d_out + (size_t)T_SEQ * DMODEL; // outputs [T, D]
    // Yout doubles as Pre scratch: phase 1 writes Pre there, the recurrence
    // consumes it, phase 3 overwrites it with the final Y. No big ws needed.
    unsigned* bar = (unsigned*)d_ws;

    hipMemsetAsync(d_ws, 0, 256, stream);  // barrier state (capture-safe)

    dim3 gemm_grid(DMODEL / 64, T_SEQ / 128);
    dim3 gemm_block(256);

    // Phase 1: Pre = X @ Wx + bh
    gemm_bias_f32_wmma<<<gemm_grid, gemm_block, 0, stream>>>(
        X, Wx, bh, Yout, T_SEQ, DMODEL, DMODEL);

    // Phase 2: sequential recurrence -> Hout
    rnn_recurrence_kernel<<<dim3(RNN_NWG), dim3(256), 0, stream>>>(
        Wh, Yout, Hout, bar);

    // Phase 3: Y = H @ Wy + by (overwrites Pre region)
    gemm_bias_f32_wmma<<<gemm_grid, gemm_block, 0, stream>>>(
        Hout, Wy, by, Yout, T_SEQ, DMODEL, DMODEL);
}